// DeformUnfold_1580547967455
// MI455X (gfx1250) — compile-verified
//
#include <hip/hip_runtime.h>
#include <cmath>

#ifndef __has_builtin
#define __has_builtin(x) 0
#endif

#if __has_builtin(__builtin_amdgcn_global_load_async_to_lds_b32)
#define USE_ASYNC_LDS 1
#else
#define USE_ASYNC_LDS 0
#endif

// ---- problem constants (from reference) ----
constexpr int Bn = 4, Cn = 64, Hn = 128, Wn = 128;
constexpr int Kn = 9;                    // 3x3
constexpr int HoN = 128, WoN = 128;
constexpr int HW = Hn * Wn;              // 16384
constexpr int PHW = HoN * WoN;           // 16384

// ---- tiling ----
constexpr int TH = 8, TW = 32;           // spatial tile per block (256 positions)
constexpr int CCH = 16;                  // channels per block
constexpr int RAD = 4;                   // learned-offset radius covered by LDS halo
constexpr int HLO = RAD + 2;             // halo rows/cols before tile start
constexpr int SH = TH + 2 * RAD + 5;     // 21 staged rows
constexpr int SW = TW + 2 * RAD + 5;     // 45 staged cols
constexpr int SWP = 48;                  // padded col stride (channel-pair interleaved)
constexpr int NTHREADS = 256;
// LDS layout: [cp 0..7][row 0..SH-1][col 0..SWP-1][c&1] -> 8*21*48*2 = 16128 floats (63 KB)
constexpr int CP_STRIDE = SH * SWP;      // in float2 units per channel-pair

__device__ __forceinline__ void stage_word(const float* g, float* l) {
#if USE_ASYNC_LDS
  // CDNA5 async copy: GLOBAL_LOAD_ASYNC_TO_LDS_B32 (tracked by ASYNCcnt)
  __builtin_amdgcn_global_load_async_to_lds_b32(
      (__attribute__((address_space(1))) int*)g,
      (__attribute__((address_space(3))) int*)l, 0, 0);
#else
  *l = __builtin_nontemporal_load(g);
#endif
}

__device__ __forceinline__ void wait_async() {
#if USE_ASYNC_LDS
#if __has_builtin(__builtin_amdgcn_s_wait_asynccnt)
  __builtin_amdgcn_s_wait_asynccnt(0);
#else
  asm volatile("s_wait_asynccnt 0" ::: "memory");
#endif
#endif
}

__global__ __launch_bounds__(NTHREADS, 1)
void deform_unfold_kernel(const float* __restrict__ x,
                          const float* __restrict__ off,
                          float* __restrict__ out) {
  __shared__ float smem[(CCH / 2) * SH * SWP * 2];

  int bid = blockIdx.x;
  const int txi = bid & 3;   bid >>= 2;   // Wo tile  (4)
  const int tyi = bid & 15;  bid >>= 4;   // Ho tile  (16)
  const int cc  = bid & 3;   bid >>= 2;   // channel chunk (4)
  const int b   = bid;                    // batch (4)
  const int cbase = cc * CCH;

  const int ty = tyi * TH, tx = txi * TW;
  const int y_lo = ty - HLO, x_lo = tx - HLO;
  const int sy0 = (y_lo > 0) ? y_lo : 0;
  const int sx0 = (x_lo > 0) ? x_lo : 0;
  const int sy1 = ((y_lo + SH) < Hn) ? (y_lo + SH) : Hn;
  const int sx1 = ((x_lo + SW) < Wn) ? (x_lo + SW) : Wn;
  const int rows = sy1 - sy0, cols = sx1 - sx0;
  const int tid = threadIdx.x;
  const int ly = tid >> 5, lx = tid & 31;  // 8 x 32 thread layout

  const float* xb = x + ((size_t)b * Cn + cbase) * HW;

  // ---- stage halo tile (16 channels) into LDS, channel-pair interleaved ----
  // Division-free 2D strided staging: lanes sweep 32 consecutive cols (coalesced),
  // the 8 thread-rows sweep the <=21 halo rows.
  for (int c = 0; c < CCH; ++c) {
    const float* xc = xb + (size_t)c * HW;
    float* lc = smem + (c >> 1) * (SH * SWP * 2) + (c & 1);
    for (int ry = ly; ry < rows; ry += TH) {
      const int yy = sy0 + ry;
      const float* xrow = xc + yy * Wn;
      float* lrow = lc + ((yy - y_lo) * SWP - x_lo) * 2;
      for (int rx = lx; rx < cols; rx += TW) {
        const int xx = sx0 + rx;
        stage_word(xrow + xx, lrow + (xx << 1));
      }
    }
  }
  wait_async();
  __syncthreads();

  // ---- per-thread output position ----
  const int ho = ty + ly, wo = tx + lx;
  const int p = ho * WoN + wo;
  const float* offb = off + (size_t)b * (2 * Kn) * PHW + p;
  float* outb = out + (size_t)b * (Cn * Kn) * PHW + p;
  const float fy = (float)(ho - 1);      // STRIDE=1, PAD=1
  const float fx = (float)(wo - 1);
  const float2* s2 = (const float2*)smem;

  for (int kyi = 0; kyi < 3; ++kyi) {
    for (int kxi = 0; kxi < 3; ++kxi) {
      const int k = kyi * 3 + kxi;
      const float oy = __builtin_nontemporal_load(offb + (size_t)(2 * k) * PHW);
      const float ox = __builtin_nontemporal_load(offb + (size_t)(2 * k + 1) * PHW);
      const float py = oy + fy + (float)kyi;
      const float px = ox + fx + (float)kxi;
      const float y0f = floorf(py), x0f = floorf(px);
      const float wy1 = py - y0f, wx1 = px - x0f;
      const float wy0 = 1.0f - wy1, wx0 = 1.0f - wx1;
      const int y0 = (int)y0f, x0 = (int)x0f;
      const int y1 = y0 + 1, x1 = x0 + 1;
      const float w00 = wy0 * wx0, w01 = wy0 * wx1;
      const float w10 = wy1 * wx0, w11 = wy1 * wx1;

      const bool vy0 = (unsigned)y0 < (unsigned)Hn, vy1 = (unsigned)y1 < (unsigned)Hn;
      const bool vx0 = (unsigned)x0 < (unsigned)Wn, vx1 = (unsigned)x1 < (unsigned)Wn;
      const bool v00 = vy0 && vx0, v01 = vy0 && vx1, v10 = vy1 && vx0, v11 = vy1 && vx1;
      const bool wy0i = (unsigned)(y0 - y_lo) < (unsigned)SH;
      const bool wy1i = (unsigned)(y1 - y_lo) < (unsigned)SH;
      const bool wx0i = (unsigned)(x0 - x_lo) < (unsigned)SW;
      const bool wx1i = (unsigned)(x1 - x_lo) < (unsigned)SW;
      const bool l00 = v00 && wy0i && wx0i;
      const bool l01 = v01 && wy0i && wx1i;
      const bool l10 = v10 && wy1i && wx0i;
      const bool l11 = v11 && wy1i && wx1i;

      // weights for the LDS path (0 when corner not staged)
      const float e00 = l00 ? w00 : 0.0f, e01 = l01 ? w01 : 0.0f;
      const float e10 = l10 ? w10 : 0.0f, e11 = l11 ? w11 : 0.0f;
      // float2 offsets inside one channel-pair plane (safe 0 when unused)
      const int i00 = l00 ? ((y0 - y_lo) * SWP + (x0 - x_lo)) : 0;
      const int i01 = l01 ? ((y0 - y_lo) * SWP + (x1 - x_lo)) : 0;
      const int i10 = l10 ? ((y1 - y_lo) * SWP + (x0 - x_lo)) : 0;
      const int i11 = l11 ? ((y1 - y_lo) * SWP + (x1 - x_lo)) : 0;
      // in-image but outside staged halo (|offset| > 4): rare exact fallback
      const bool rare = (v00 && !l00) || (v01 && !l01) || (v10 && !l10) || (v11 && !l11);

#pragma unroll
      for (int cp = 0; cp < CCH / 2; ++cp) {
        const int cpo = cp * CP_STRIDE;
        const float2 a00 = s2[cpo + i00];
        const float2 a01 = s2[cpo + i01];
        const float2 a10 = s2[cpo + i10];
        const float2 a11 = s2[cpo + i11];
        float r0 = e00 * a00.x + e01 * a01.x + e10 * a10.x + e11 * a11.x;
        float r1 = e00 * a00.y + e01 * a01.y + e10 * a10.y + e11 * a11.y;
        if (rare) {
          const float* xc0 = xb + (size_t)(2 * cp) * HW;
          const float* xc1 = xc0 + HW;
          if (v00 && !l00) { const int g = y0 * Wn + x0; r0 += w00 * xc0[g]; r1 += w00 * xc1[g]; }
          if (v01 && !l01) { const int g = y0 * Wn + x1; r0 += w01 * xc0[g]; r1 += w01 * xc1[g]; }
          if (v10 && !l10) { const int g = y1 * Wn + x0; r0 += w10 * xc0[g]; r1 += w10 * xc1[g]; }
          if (v11 && !l11) { const int g = y1 * Wn + x1; r0 += w11 * xc0[g]; r1 += w11 * xc1[g]; }
        }
        // streaming output: non-temporal so the 151MB result doesn't evict L2-resident x
        __builtin_nontemporal_store(r0, outb + (size_t)((cbase + 2 * cp) * Kn + k) * PHW);
        __builtin_nontemporal_store(r1, outb + (size_t)((cbase + 2 * cp + 1) * Kn + k) * PHW);
      }
    }
  }
}

extern "C" void kernel_launch(void* const* d_in, const int* in_sizes, int n_in,
                              void* d_out, int out_size, void* d_ws, size_t ws_size,
                              hipStream_t stream) {
  (void)in_sizes; (void)n_in; (void)out_size; (void)d_ws; (void)ws_size;
  const float* x = (const float*)d_in[0];     // (4, 64, 128, 128) f32
  const float* off = (const float*)d_in[1];   // (4, 18, 128, 128) f32
  float* out = (float*)d_out;                 // (4, 576, 16384) f32
  const int grid = Bn * (HoN / TH) * (WoN / TW) * (Cn / CCH);  // 4*16*4*4 = 1024
  deform_unfold_kernel<<<grid, NTHREADS, 0, stream>>>(x, off, out);
}